// ALLGAT_91182155694423
// MI455X (gfx1250) — compile-verified
//
#include <hip/hip_runtime.h>
#include <hip/hip_bf16.h>
#include <math.h>

#define HEADS 8
#define DHEAD 16
#define DIM   128  // HEADS * DHEAD == IN_DIM

typedef __attribute__((ext_vector_type(2))) float v2f;
typedef __attribute__((ext_vector_type(8))) float v8f;

// ---------------------------------------------------------------- fill
__global__ void gat_fill_kernel(float* __restrict__ p, float v, int n) {
    int i = blockIdx.x * blockDim.x + threadIdx.x;
    if (i < n) p[i] = v;
}

// ---------------------------------------------------------------- projection (WMMA f32 16x16x4)
// z[n, h*16+k] = sum_d hin[n,d] * W[h,d,k]   (W stored [HEADS,128,16])
// One wave per 16-row x 16-col tile; col tile == one head.
__global__ __launch_bounds__(32) void gat_proj_wmma_kernel(
    const float* __restrict__ hin, const float* __restrict__ W,
    float* __restrict__ z, int nrows)
{
    const int lane = threadIdx.x;        // 0..31, wave32
    const int m    = lane & 15;
    const int koff = (lane >> 4) << 1;   // lanes 0-15 -> K+0/K+1, lanes 16-31 -> K+2/K+3
    const int row0 = blockIdx.x * 16;
    const int head = blockIdx.y;         // 0..7

    int arow_idx = row0 + m;
    if (arow_idx >= nrows) arow_idx = nrows - 1;     // clamp read, keep EXEC all-ones
    const float* arow = hin + (size_t)arow_idx * DIM;
    // B[d][n] with n = lane&15:  W[head*2048 + d*16 + n]
    const float* wb = W + (size_t)head * (DIM * DHEAD) + m;

    v8f acc = {};
#pragma unroll
    for (int k = 0; k < DIM; k += 4) {
        v2f a;
        a.x = arow[k + koff];
        a.y = arow[k + koff + 1];
        v2f b;
        b.x = wb[(size_t)(k + koff) * DHEAD];
        b.y = wb[(size_t)(k + koff + 1) * DHEAD];
        acc = __builtin_amdgcn_wmma_f32_16x16x4_f32(
            /*neg_a=*/false, a, /*neg_b=*/false, b,
            /*c_mod=*/(short)0, acc, /*reuse_a=*/false, /*reuse_b=*/false);
    }

    // D layout: VGPR r -> row r (lanes 0-15) / row 8+r (lanes 16-31), col = lane&15
    const int rbase = (lane >> 4) << 3;  // 0 or 8
#pragma unroll
    for (int r = 0; r < 8; ++r) {
        int row = row0 + rbase + r;
        if (row < nrows)
            z[(size_t)row * DIM + head * DHEAD + m] = acc[r];
    }
}

// ---------------------------------------------------------------- per-(node,head) attention scores
__global__ void gat_scores_kernel(const float* __restrict__ z,
                                  const float* __restrict__ a_src,
                                  const float* __restrict__ a_dst,
                                  float* __restrict__ s_src,
                                  float* __restrict__ s_dst, int n_nodes)
{
    int t = blockIdx.x * blockDim.x + threadIdx.x;
    if (t >= n_nodes * HEADS) return;
    int h = t & (HEADS - 1);
    int n = t >> 3;
    const float* zp = z + (size_t)n * DIM + h * DHEAD;
    const float* as = a_src + h * DHEAD;
    const float* ad = a_dst + h * DHEAD;
    float ss = 0.f, sd = 0.f;
#pragma unroll
    for (int k = 0; k < DHEAD; ++k) {
        float v = zp[k];
        ss += v * as[k];
        sd += v * ad[k];
    }
    s_src[t] = ss;
    s_dst[t] = sd;
}

// ---------------------------------------------------------------- float atomic max (native int/uint path)
__device__ inline void atomicMaxFloat(float* addr, float val) {
    if (val >= 0.0f) {
        atomicMax((int*)addr, __float_as_int(val));
    } else {
        atomicMin((unsigned int*)addr, (unsigned int)__float_as_int(val));
    }
}

// ---------------------------------------------------------------- segment max over dst
__global__ void gat_edge_max_kernel(const int* __restrict__ src, const int* __restrict__ dst,
                                    const float* __restrict__ s_src, const float* __restrict__ s_dst,
                                    float* __restrict__ emax, int n_edges)
{
    int t = blockIdx.x * blockDim.x + threadIdx.x;
    if (t >= n_edges * HEADS) return;
    int h = t & (HEADS - 1);
    int e = t >> 3;
    int s = src[e];
    int d = dst[e];
    float ev = s_src[s * HEADS + h] + s_dst[d * HEADS + h];
    ev = ev > 0.f ? ev : 0.01f * ev;                 // leaky_relu
    atomicMaxFloat(&emax[d * HEADS + h], ev);
}

// ---------------------------------------------------------------- softmax weights + message scatter
__global__ void gat_edge_accum_kernel(const int* __restrict__ src, const int* __restrict__ dst,
                                      const float* __restrict__ s_src, const float* __restrict__ s_dst,
                                      const float* __restrict__ emax, const float* __restrict__ z,
                                      float* __restrict__ denom, float* __restrict__ num, int n_edges)
{
    int t = blockIdx.x * blockDim.x + threadIdx.x;
    if (t >= n_edges * HEADS) return;
    int h = t & (HEADS - 1);
    int e = t >> 3;
    int s = src[e];
    int d = dst[e];
    float ev = s_src[s * HEADS + h] + s_dst[d * HEADS + h];
    ev = ev > 0.f ? ev : 0.01f * ev;
    float w = __expf(ev - emax[d * HEADS + h]);
    atomicAdd(&denom[d * HEADS + h], w);

    const float4* zp = (const float4*)(z + (size_t)s * DIM + h * DHEAD);  // 16B aligned
    float* np = num + (size_t)d * DIM + h * DHEAD;
#pragma unroll
    for (int q = 0; q < 4; ++q) {
        float4 zv = zp[q];
        atomicAdd(&np[q * 4 + 0], w * zv.x);
        atomicAdd(&np[q * 4 + 1], w * zv.y);
        atomicAdd(&np[q * 4 + 2], w * zv.z);
        atomicAdd(&np[q * 4 + 3], w * zv.w);
    }
}

// ---------------------------------------------------------------- normalize (+ optional ELU)
__global__ void gat_finalize_kernel(const float* __restrict__ num, const float* __restrict__ denom,
                                    float* __restrict__ out, int n_nodes, int apply_elu)
{
    int t = blockIdx.x * blockDim.x + threadIdx.x;
    if (t >= n_nodes * DIM) return;
    int n = t >> 7;             // /128
    int h = (t & (DIM - 1)) >> 4;
    float dn = denom[n * HEADS + h];
    dn = (dn == 0.f) ? 1.f : dn;
    float v = num[t] / dn;
    if (apply_elu) v = v > 0.f ? v : expm1f(v);
    out[t] = v;
}

// ---------------------------------------------------------------- launch
extern "C" void kernel_launch(void* const* d_in, const int* in_sizes, int n_in,
                              void* d_out, int out_size, void* d_ws, size_t ws_size,
                              hipStream_t stream) {
    const float* x   = (const float*)d_in[0];
    const int*   src = (const int*)  d_in[1];
    const int*   dst = (const int*)  d_in[2];
    const float* W1  = (const float*)d_in[3];
    const float* a1s = (const float*)d_in[4];
    const float* a1d = (const float*)d_in[5];
    const float* W2  = (const float*)d_in[6];
    const float* a2s = (const float*)d_in[7];
    const float* a2d = (const float*)d_in[8];
    float* out = (float*)d_out;

    const int N = in_sizes[0] / DIM;
    const int E = in_sizes[1];

    float* ws    = (float*)d_ws;
    float* z     = ws;                      // 128*N
    float* ssrc  = z    + (size_t)DIM * N;  // 8*N
    float* sdst  = ssrc + (size_t)HEADS * N;
    float* emax  = sdst + (size_t)HEADS * N;
    float* denom = emax + (size_t)HEADS * N;
    float* num   = denom + (size_t)HEADS * N;  // 128*N (doubles as h1 after layer-1 finalize)

    const int NH = N * HEADS;
    const int NC = N * DIM;
    const int EH = E * HEADS;
    auto cdiv = [](int a, int b) { return (a + b - 1) / b; };

    for (int layer = 0; layer < 2; ++layer) {
        const float* hin = layer ? num : x;      // layer-2 input = ELU'd layer-1 output (in num)
        const float* W   = layer ? W2 : W1;
        const float* as  = layer ? a2s : a1s;
        const float* ad  = layer ? a2d : a1d;
        float*       acc = layer ? out : num;    // accumulation target

        gat_fill_kernel<<<cdiv(NH, 256), 256, 0, stream>>>(emax, -INFINITY, NH);
        gat_fill_kernel<<<cdiv(NH, 256), 256, 0, stream>>>(denom, 0.f, NH);
        gat_fill_kernel<<<cdiv(NC, 256), 256, 0, stream>>>(acc, 0.f, NC);

        gat_proj_wmma_kernel<<<dim3(cdiv(N, 16), HEADS), 32, 0, stream>>>(hin, W, z, N);
        gat_scores_kernel<<<cdiv(NH, 256), 256, 0, stream>>>(z, as, ad, ssrc, sdst, N);
        gat_edge_max_kernel<<<cdiv(EH, 256), 256, 0, stream>>>(src, dst, ssrc, sdst, emax, E);
        gat_edge_accum_kernel<<<cdiv(EH, 256), 256, 0, stream>>>(src, dst, ssrc, sdst, emax, z,
                                                                 denom, acc, E);
        gat_finalize_kernel<<<cdiv(NC, 256), 256, 0, stream>>>(acc, denom, layer ? out : num,
                                                               N, layer == 0 ? 1 : 0);
    }
}